// HebbyRNN_18227841204731
// MI455X (gfx1250) — compile-verified
//
#include <hip/hip_runtime.h>
#include <hip/hip_bf16.h>

// HebbyRNN forward for MI455X (gfx1250, wave32).
// Bandwidth-bound batched matvec chain: ~1.02 GB of fp32 weights streamed once
// (~44 us floor at 23.3 TB/s). K-reduction runs in the matrix pipe with
// V_WMMA_F32_16X16X4_F32 (full fp32). Weights feed the A fragment (documented
// layout). The input vector is replicated into ALL columns of B, so every
// column of D holds the same matvec result: no lane masking, no EXEC
// divergence, uniform broadcast LDS loads.

#define BATCH   64
#define IN_SIZE 128
#define HID     1024
#define OUTSZ   128
#define INNER   1152   // IN_SIZE + HID

typedef __attribute__((ext_vector_type(2))) float v2f;
typedef __attribute__((ext_vector_type(8))) float v8f;

// ---------------------------------------------------------------------------
// combined = concat(x, hidden) : [B, INNER]
// ---------------------------------------------------------------------------
__global__ __launch_bounds__(256) void concat_kernel(
    const float* __restrict__ x, const float* __restrict__ h,
    float* __restrict__ comb)
{
    int i = blockIdx.x * blockDim.x + threadIdx.x;
    if (i >= BATCH * INNER) return;
    int b = i / INNER;
    int j = i - b * INNER;
    comb[i] = (j < IN_SIZE) ? x[b * IN_SIZE + j] : h[b * HID + (j - IN_SIZE)];
}

// ---------------------------------------------------------------------------
// vout[b, m] = act( sum_k W[b, m, k] * vin[b, k] + bias[m] ),  K == INNER.
// One wave per (sample b = blockIdx.y, 32-row chunk = blockIdx.x).
// Two 16x16 accumulators share one B-fragment stream.
//   A (16x4 fp32, documented): lane L<16: M=L, K={0,1}; lane L>=16: M=L-16, K={2,3}
//   B: every lane carries {v[kk+khalf], v[kk+khalf+1]} -> all columns of B are
//      the input vector, so every column of D is the matvec result.
//   D: lane 0: rows 0..7 in acc[0..7]; lane 16: rows 8..15.
// ---------------------------------------------------------------------------
__global__ __launch_bounds__(32) void hebb_matvec_wmma(
    const float* __restrict__ W, const float* __restrict__ bias,
    const float* __restrict__ vin, float* __restrict__ vout,
    int M, int do_relu)
{
    __shared__ __align__(16) float sv[INNER];

    const int b       = blockIdx.y;
    const int rowBase = blockIdx.x << 5;     // 32 rows per wave
    const int lane    = threadIdx.x;         // full wave, EXEC all 1s

    // Stage this sample's input vector into LDS (coalesced b128 loads).
    const float* v = vin + (size_t)b * INNER;
    for (int i = lane * 4; i < INNER; i += 32 * 4) {
        *reinterpret_cast<float4*>(&sv[i]) =
            *reinterpret_cast<const float4*>(&v[i]);
    }
    __syncthreads();

    const int khalf = (lane >> 4) << 1;      // 0 (lanes<16) or 2 (lanes>=16)
    const int row0  = rowBase + (lane & 15);
    const float* wr0 = W + ((size_t)b * M + row0) * INNER + khalf;
    const float* wr1 = wr0 + (size_t)16 * INNER;

    v8f acc0 = {};
    v8f acc1 = {};

    for (int k0 = 0; k0 < INNER; k0 += 16) {
        v2f a0[4], a1[4], bf[4];
        // Batch all loads first: 8 global_load_b64 + 4 broadcast ds_load_b64
        // in flight before any WMMA waits on them.
#pragma unroll
        for (int u = 0; u < 4; ++u) {
            const int kk = k0 + (u << 2);
            a0[u] = *reinterpret_cast<const v2f*>(wr0 + kk);
            a1[u] = *reinterpret_cast<const v2f*>(wr1 + kk);
            bf[u] = *reinterpret_cast<const v2f*>(&sv[kk + khalf]);
        }
#pragma unroll
        for (int u = 0; u < 4; ++u) {
            acc0 = __builtin_amdgcn_wmma_f32_16x16x4_f32(
                false, a0[u], false, bf[u], (short)0, acc0, false, false);
            acc1 = __builtin_amdgcn_wmma_f32_16x16x4_f32(
                false, a1[u], false, bf[u], (short)0, acc1, false, false);
        }
    }

    // Every D column is the result; read column from lanes 0 and 16.
    // lane 0:  acc0 -> rows rowBase+0..7,   acc1 -> rows rowBase+16..23
    // lane 16: acc0 -> rows rowBase+8..15,  acc1 -> rows rowBase+24..31
    if ((lane & 15) == 0) {
        const int off = (lane >> 4) << 3;    // 0 or 8
        const int m0  = rowBase + off;
        float* o = vout + (size_t)b * M + m0;
#pragma unroll
        for (int j = 0; j < 8; ++j) {
            float v0 = acc0[j] + bias[m0 + j];
            float v1 = acc1[j] + bias[m0 + 16 + j];
            if (do_relu) { v0 = fmaxf(v0, 0.0f); v1 = fmaxf(v1, 0.0f); }
            o[j]      = v0;
            o[16 + j] = v1;
        }
    }
}

// ---------------------------------------------------------------------------
// Row-wise log-softmax over OUTSZ=128 logits; one wave per sample (wave32).
// ---------------------------------------------------------------------------
__global__ __launch_bounds__(32) void logsoftmax_kernel(
    const float* __restrict__ logits, float* __restrict__ out)
{
    const int b    = blockIdx.x;
    const int lane = threadIdx.x;
    const float* in = logits + (size_t)b * OUTSZ;

    float v[4];
    float mx = -3.402823466e+38f;
#pragma unroll
    for (int j = 0; j < 4; ++j) {
        v[j] = in[lane + 32 * j];
        mx = fmaxf(mx, v[j]);
    }
#pragma unroll
    for (int o = 16; o > 0; o >>= 1)
        mx = fmaxf(mx, __shfl_xor(mx, o, 32));

    float s = 0.0f;
#pragma unroll
    for (int j = 0; j < 4; ++j) s += expf(v[j] - mx);
#pragma unroll
    for (int o = 16; o > 0; o >>= 1)
        s += __shfl_xor(s, o, 32);

    const float lse = mx + logf(s);
    float* ob = out + (size_t)b * OUTSZ;
#pragma unroll
    for (int j = 0; j < 4; ++j) ob[lane + 32 * j] = v[j] - lse;
}

// ---------------------------------------------------------------------------
extern "C" void kernel_launch(void* const* d_in, const int* in_sizes, int n_in,
                              void* d_out, int out_size, void* d_ws, size_t ws_size,
                              hipStream_t stream) {
    (void)in_sizes; (void)n_in; (void)out_size; (void)ws_size;

    const float* x      = (const float*)d_in[0];
    const float* hidden = (const float*)d_in[1];
    const float* cw_l0  = (const float*)d_in[2];
    const float* b_l0   = (const float*)d_in[3];
    const float* cw_l1  = (const float*)d_in[4];
    const float* b_l1   = (const float*)d_in[5];
    const float* cw_i2h = (const float*)d_in[6];
    const float* b_i2h  = (const float*)d_in[7];
    const float* cw_i2o = (const float*)d_in[8];
    const float* b_i2o  = (const float*)d_in[9];

    float* out_logsm = (float*)d_out;                 // [B, OUTSZ]
    float* out_hid   = (float*)d_out + BATCH * OUTSZ; // [B, HID]

    float* comb   = (float*)d_ws;                     // [B, INNER]
    float* c1     = comb + BATCH * INNER;             // [B, INNER]
    float* logits = c1 + BATCH * INNER;               // [B, OUTSZ]

    // combined = concat(x, hidden)
    concat_kernel<<<(BATCH * INNER + 255) / 256, 256, 0, stream>>>(x, hidden, comb);

    // Layer 0: c1 = relu(cw_l0 @ comb + b_l0)
    dim3 gL(INNER / 32, BATCH);
    hebb_matvec_wmma<<<gL, 32, 0, stream>>>(cw_l0, b_l0, comb, c1, INNER, 1);
    // Layer 1: c2 = relu(cw_l1 @ c1 + b_l1)   (c2 reuses comb buffer)
    hebb_matvec_wmma<<<gL, 32, 0, stream>>>(cw_l1, b_l1, c1, comb, INNER, 1);

    // new_hidden = cw_i2h @ c2 + b_i2h  -> second half of d_out
    dim3 gH(HID / 32, BATCH);
    hebb_matvec_wmma<<<gH, 32, 0, stream>>>(cw_i2h, b_i2h, comb, out_hid, HID, 0);

    // logits = cw_i2o @ c2 + b_i2o
    dim3 gO(OUTSZ / 32, BATCH);
    hebb_matvec_wmma<<<gO, 32, 0, stream>>>(cw_i2o, b_i2o, comb, logits, OUTSZ, 0);

    // output = log_softmax(logits) -> first half of d_out
    logsoftmax_kernel<<<BATCH, 32, 0, stream>>>(logits, out_logsm);
}